// NearDiagNonNegMixIO_47974784697007
// MI455X (gfx1250) — compile-verified
//
#include <hip/hip_runtime.h>
#include <cmath>

// ---- problem constants (match reference) ----
#define IN_DIM  256
#define OUT_DIM 256
#define HW      65536           // 256*256 pixels, contiguous per (b, channel)
#define BATCH   8
#define BAND    4               // |o - i| <= 4 from mask construction (bw=2)
#define EPS     1e-3f

typedef __attribute__((ext_vector_type(2))) float v2f;
typedef __attribute__((ext_vector_type(8))) float v8f;

__device__ __forceinline__ float softplus_f32(float p) {
    // jax.nn.softplus(x) == max(x,0) + log1p(exp(-|x|))
    return fmaxf(p, 0.0f) + log1pf(expf(-fabsf(p)));
}

__device__ __forceinline__ v8f wmma_f32_k4(v2f a, v2f b, v8f c) {
    return __builtin_amdgcn_wmma_f32_16x16x4_f32(
        /*neg_a=*/false, a, /*neg_b=*/false, b,
        /*c_mod=*/(short)0, c, /*reuse_a=*/false, /*reuse_b=*/false);
}

// Banded per-pixel channel mix via V_WMMA_F32_16X16X4_F32.
// Each wave: 16(M=o) x 32(N=hw) output per iteration, band K = 24 -> 12 WMMAs.
__global__ void __launch_bounds__(256)
near_diag_mix_wmma(const float* __restrict__ x,
                   const float* __restrict__ P,
                   const float* __restrict__ mask,
                   const float* __restrict__ eye,
                   float* __restrict__ out)
{
    const int lane  = threadIdx.x & 31;
    const int wave  = threadIdx.x >> 5;
    const int m     = blockIdx.y;              // M block: o in [16m, 16m+16)
    const int b     = blockIdx.z;

    const int mrow  = lane & 15;               // A: M row held by lane; B/D: N column index
    const int kadd  = (lane >> 4) << 1;        // A/B: upper half-wave holds K+2, K+3
    const int obase = m * 16;
    const int kbase0 = obase - BAND;           // first band K, multiple of 4

    const float* __restrict__ xb = x   + (size_t)b * IN_DIM  * HW;
    float*       __restrict__ ob = out + (size_t)b * OUT_DIM * HW;

    const int ntiles  = HW / 16;               // 4096 column tiles
    const int tstride = gridDim.x * 8;         // waves spanning N (8 waves/block)

    const int o = obase + mrow;

    if (m > 0 && m < (OUT_DIM / 16) - 1) {
        // ---------- interior fast path: band fully in range, constant offsets ----------
        v2f afrag[6];
        #pragma unroll
        for (int kc = 0; kc < 6; ++kc) {
            const int i0  = kbase0 + kc * 4 + kadd;
            const int idx = o * IN_DIM + i0;
            afrag[kc].x = softplus_f32(P[idx])     * mask[idx]     + EPS * eye[idx];
            afrag[kc].y = softplus_f32(P[idx + 1]) * mask[idx + 1] + EPS * eye[idx + 1];
        }

        // per-lane base: row (kbase0 + kadd), this lane's column
        const float* __restrict__ bbase = xb + (ptrdiff_t)(kbase0 + kadd) * HW;

        for (int t = (blockIdx.x * 8 + wave) * 2; t < ntiles; t += tstride * 2) {
            const float* __restrict__ p = bbase + t * 16 + mrow;

            v8f acc0 = {}, acc1 = {};
            #pragma unroll
            for (int kc = 0; kc < 6; ++kc) {
                v2f b0, b1;
                b0.x = p[kc * 4 * HW];            // immediate offsets, no VALU addr math
                b0.y = p[kc * 4 * HW + HW];
                b1.x = p[kc * 4 * HW + 16];
                b1.y = p[kc * 4 * HW + HW + 16];
                acc0 = wmma_f32_k4(afrag[kc], b0, acc0);
                acc1 = wmma_f32_k4(afrag[kc], b1, acc1);
            }

            // D layout: VGPR v, lane l -> M = v + (l>=16 ? 8 : 0), N = l%16
            float* __restrict__ d = ob + (size_t)(obase + ((lane >> 4) << 3)) * HW
                                       + (size_t)(t * 16 + mrow);
            #pragma unroll
            for (int v = 0; v < 8; ++v) {
                __builtin_nontemporal_store(acc0[v], d + (size_t)v * HW);
                __builtin_nontemporal_store(acc1[v], d + (size_t)v * HW + 16);
            }
        }
    } else {
        // ---------- edge path (m == 0 or last): clamped rows, zeroed A outside ----------
        v2f afrag[6];
        size_t rowoff0[6], rowoff1[6];
        #pragma unroll
        for (int kc = 0; kc < 6; ++kc) {
            const int i0 = kbase0 + kc * 4 + kadd;
            const int i1 = i0 + 1;
            float w0 = 0.0f, w1 = 0.0f;
            if (i0 >= 0 && i0 < IN_DIM) {
                const int idx = o * IN_DIM + i0;
                w0 = softplus_f32(P[idx]) * mask[idx] + EPS * eye[idx];
            }
            if (i1 >= 0 && i1 < IN_DIM) {
                const int idx = o * IN_DIM + i1;
                w1 = softplus_f32(P[idx]) * mask[idx] + EPS * eye[idx];
            }
            afrag[kc].x = w0;
            afrag[kc].y = w1;
            const int c0 = i0 < 0 ? 0 : (i0 >= IN_DIM ? IN_DIM - 1 : i0);
            const int c1 = i1 < 0 ? 0 : (i1 >= IN_DIM ? IN_DIM - 1 : i1);
            rowoff0[kc] = (size_t)c0 * HW;
            rowoff1[kc] = (size_t)c1 * HW;
        }

        for (int t = (blockIdx.x * 8 + wave) * 2; t < ntiles; t += tstride * 2) {
            #pragma unroll
            for (int u = 0; u < 2; ++u) {
                const int ncol = (t + u) * 16 + mrow;
                v8f acc = {};
                #pragma unroll
                for (int kc = 0; kc < 6; ++kc) {
                    v2f bf;
                    bf.x = xb[rowoff0[kc] + ncol];
                    bf.y = xb[rowoff1[kc] + ncol];
                    acc = wmma_f32_k4(afrag[kc], bf, acc);
                }
                float* __restrict__ d = ob + (size_t)(obase + ((lane >> 4) << 3)) * HW
                                           + (size_t)ncol;
                #pragma unroll
                for (int v = 0; v < 8; ++v)
                    __builtin_nontemporal_store(acc[v], d + (size_t)v * HW);
            }
        }
    }
}

extern "C" void kernel_launch(void* const* d_in, const int* in_sizes, int n_in,
                              void* d_out, int out_size, void* d_ws, size_t ws_size,
                              hipStream_t stream) {
    (void)in_sizes; (void)n_in; (void)out_size; (void)d_ws; (void)ws_size;
    const float* x    = (const float*)d_in[0];
    const float* P    = (const float*)d_in[1];
    const float* mask = (const float*)d_in[2];
    const float* eye  = (const float*)d_in[3];
    float* out        = (float*)d_out;

    dim3 grid(64, OUT_DIM / 16, BATCH);   // (N superchunks, M blocks, batch)
    dim3 block(256);                       // 8 wave32 waves
    near_diag_mix_wmma<<<grid, block, 0, stream>>>(x, P, mask, eye, out);
}